// DeconvProp_S1_29162827940276
// MI455X (gfx1250) — compile-verified
//
#include <hip/hip_runtime.h>
#include <hip/hip_bf16.h>

// ---------------------------------------------------------------------------
// Problem constants (match reference)
// ---------------------------------------------------------------------------
#define N_GENE   50000
#define N_CT     64
#define N_REF    16
#define N_SAMPLE 1024

#define N_Y0 (N_CT * N_GENE)          // 3,200,000
#define N_Y1 (N_GENE * N_CT)          // 3,200,000
#define N_OUT (N_SAMPLE * N_GENE)     // 51,200,000

typedef float v2f __attribute__((ext_vector_type(2)));
typedef float v4f __attribute__((ext_vector_type(4)));
typedef float v8f __attribute__((ext_vector_type(8)));

// ---------------------------------------------------------------------------
// Kernel 1: y0[i] = sum_{r<16} x[16i+r] * w_ref[(i%64)*16 + r]
// One thread per y0 element; 4x b128 loads of x + 4x b128 of w_ref (L2 hot).
// ---------------------------------------------------------------------------
__global__ void __launch_bounds__(256)
k_refcomb_sum(const float* __restrict__ x,
              const float* __restrict__ w_ref,
              float* __restrict__ y0)
{
    const int i = blockIdx.x * 256 + threadIdx.x;          // 0 .. 3.2M-1 exact
    const size_t xb = (size_t)i * 16;                      // 64B aligned
    const int wb = (i & (N_CT - 1)) * 16;                  // 64B aligned

    const v4f* xp = (const v4f*)(x + xb);
    const v4f* wp = (const v4f*)(w_ref + wb);

    v4f acc = xp[0] * wp[0];
    acc += xp[1] * wp[1];
    acc += xp[2] * wp[2];
    acc += xp[3] * wp[3];
    y0[i] = acc.x + acc.y + acc.z + acc.w;
}

// ---------------------------------------------------------------------------
// Kernel 2: y1[g*64+c] = y0[idx[c*N_GENE+g]] * w_ct[c];  y2 = y1 * w_stretch[g]
// Tile: 100 genes x 64 celltypes per block, staged via LDS so idx reads are
// gene-contiguous (coalesced) and y1/y2 writes are fully contiguous.
// ---------------------------------------------------------------------------
#define TILE_G 100   // divides 50000 -> 500 blocks; 6400 elems = 25 iters of 256

__global__ void __launch_bounds__(256)
k_gather_scale(const float* __restrict__ y0,
               const int*   __restrict__ idx,
               const float* __restrict__ w_ct,
               const float* __restrict__ w_stretch,
               float* __restrict__ y1,
               float* __restrict__ y2)
{
    __shared__ float tile[TILE_G * 65];   // stride 65 -> odd stride, no bank conflicts
    const int t  = threadIdx.x;
    const int g0 = blockIdx.x * TILE_G;

    // Phase 1: gather, c-major / gene-fast (coalesced idx reads)
    #pragma unroll 5
    for (int it = 0; it < (TILE_G * N_CT) / 256; ++it) {
        const int e  = it * 256 + t;       // e = c*TILE_G + gi
        const int c  = e / TILE_G;
        const int gi = e - c * TILE_G;
        const int id = idx[c * N_GENE + (g0 + gi)];
        tile[gi * 65 + c] = y0[id] * w_ct[c];
    }
    __syncthreads();

    // Phase 2: write gene-major (fully coalesced y1/y2 stores)
    #pragma unroll 5
    for (int it = 0; it < (TILE_G * N_CT) / 256; ++it) {
        const int e  = it * 256 + t;       // e = gi*64 + c
        const int gi = e >> 6;
        const int c  = e & 63;
        const float v1 = tile[gi * 65 + c];
        const int p = (g0 + gi) * N_CT + c;
        y1[p] = v1;
        y2[p] = v1 * w_stretch[g0 + gi];
    }
}

// ---------------------------------------------------------------------------
// Kernel 3: out[s,g] = sum_k y2[g*64+k] * conv_w[s*64+k]
// GEMM M=50000, N=1024, K=64 with V_WMMA_F32_16X16X4_F32.
// One wave -> 16 genes x 64 samples (4 accumulator tiles, shared A operand),
// 16 K-steps x 4 WMMAs = 64 wmma per wave. 50000 waves exactly (EXEC all-1s).
//
// VGPR layouts per CDNA5 ISA §7.12.2 (f32):
//   A 16x4 : lane%16 = M(gene), float2 = K = 2*(lane/16) + {0,1}
//   B 4x16 : lane%16 = N(sample), vgpr v + 2*(lane/16) = K  -> float2 of K
//   D 16x16: lane%16 = N, vgpr v -> M = v + 8*(lane/16)     -> gene-contiguous
// ---------------------------------------------------------------------------
__global__ void __launch_bounds__(256)
k_gemm_wmma(const float* __restrict__ yk,     // y2 viewed as [N_GENE][64]
            const float* __restrict__ conv,   // [N_SAMPLE][64]
            float* __restrict__ out)          // [N_SAMPLE][N_GENE]
{
    const int lane = threadIdx.x & 31;
    const int wave = blockIdx.x * 8 + (threadIdx.x >> 5);  // 0..49999
    const int g0 = (wave >> 4) * 16;                       // gene tile base
    const int s0 = (wave & 15) * 64;                       // sample block base
    const int ln = lane & 15;
    const int lh = lane >> 4;

    v8f acc0 = {}, acc1 = {}, acc2 = {}, acc3 = {};

    const float* aBase = yk   + (size_t)(g0 + ln) * 64 + 2 * lh;
    const float* bBase = conv + (size_t)(s0 + ln) * 64 + 2 * lh;

    #pragma unroll
    for (int ks = 0; ks < 16; ++ks) {
        const int kb = ks * 4;
        const v2f a  = *(const v2f*)(aBase + kb);
        const v2f b0 = *(const v2f*)(bBase + kb);
        const v2f b1 = *(const v2f*)(bBase + 16 * 64 + kb);
        const v2f b2 = *(const v2f*)(bBase + 32 * 64 + kb);
        const v2f b3 = *(const v2f*)(bBase + 48 * 64 + kb);
        acc0 = __builtin_amdgcn_wmma_f32_16x16x4_f32(false, a, false, b0, (short)0, acc0, false, false);
        acc1 = __builtin_amdgcn_wmma_f32_16x16x4_f32(false, a, false, b1, (short)0, acc1, false, false);
        acc2 = __builtin_amdgcn_wmma_f32_16x16x4_f32(false, a, false, b2, (short)0, acc2, false, false);
        acc3 = __builtin_amdgcn_wmma_f32_16x16x4_f32(false, a, false, b3, (short)0, acc3, false, false);
    }

    // Store: lane holds genes g0+8*lh+{0..7} for sample s0+j*16+ln -> 2x b128
    const size_t gcol = (size_t)g0 + 8 * lh;
    {
        float* d = out + (size_t)(s0 +  0 + ln) * N_GENE + gcol;
        *(v4f*)(d)     = __builtin_shufflevector(acc0, acc0, 0, 1, 2, 3);
        *(v4f*)(d + 4) = __builtin_shufflevector(acc0, acc0, 4, 5, 6, 7);
    }
    {
        float* d = out + (size_t)(s0 + 16 + ln) * N_GENE + gcol;
        *(v4f*)(d)     = __builtin_shufflevector(acc1, acc1, 0, 1, 2, 3);
        *(v4f*)(d + 4) = __builtin_shufflevector(acc1, acc1, 4, 5, 6, 7);
    }
    {
        float* d = out + (size_t)(s0 + 32 + ln) * N_GENE + gcol;
        *(v4f*)(d)     = __builtin_shufflevector(acc2, acc2, 0, 1, 2, 3);
        *(v4f*)(d + 4) = __builtin_shufflevector(acc2, acc2, 4, 5, 6, 7);
    }
    {
        float* d = out + (size_t)(s0 + 48 + ln) * N_GENE + gcol;
        *(v4f*)(d)     = __builtin_shufflevector(acc3, acc3, 0, 1, 2, 3);
        *(v4f*)(d + 4) = __builtin_shufflevector(acc3, acc3, 4, 5, 6, 7);
    }
}

// ---------------------------------------------------------------------------
// Launcher.  d_out layout (tuple concat): out | y0 | y1 | y2
// ---------------------------------------------------------------------------
extern "C" void kernel_launch(void* const* d_in, const int* in_sizes, int n_in,
                              void* d_out, int out_size, void* d_ws, size_t ws_size,
                              hipStream_t stream)
{
    const float* x         = (const float*)d_in[0];
    const int*   idx       = (const int*)  d_in[1];
    const float* w_ref     = (const float*)d_in[2];
    const float* w_ct      = (const float*)d_in[3];
    const float* w_stretch = (const float*)d_in[4];
    const float* conv_w    = (const float*)d_in[5];

    float* out = (float*)d_out;
    float* y0  = out + (size_t)N_OUT;
    float* y1  = y0 + (size_t)N_Y0;
    float* y2  = y1 + (size_t)N_Y1;

    // K1: 3.2M y0 elements, 256 threads/block -> 12500 blocks (exact)
    k_refcomb_sum<<<N_Y0 / 256, 256, 0, stream>>>(x, w_ref, y0);

    // K2: 500 blocks of 100-gene tiles (exact)
    k_gather_scale<<<N_GENE / TILE_G, 256, 0, stream>>>(y0, idx, w_ct, w_stretch, y1, y2);

    // K3: 3125 gene-tiles * 16 sample-blocks = 50000 waves, 8 waves/block -> 6250 blocks
    k_gemm_wmma<<<(N_GENE / 16) * 16 / 8, 256, 0, stream>>>(y2, conv_w, out);
}